// motif_dist_57372173140490
// MI455X (gfx1250) — compile-verified
//
#include <hip/hip_runtime.h>
#include <math.h>

typedef __attribute__((ext_vector_type(2))) float v2f;
typedef __attribute__((ext_vector_type(8))) float v8f;

#define Bn 2
#define Cn 128
#define Hn 192
#define Wn 192
#define Gn 16
#define CG 8
#define Hm 190
#define Wm 190

// ---------------------------------------------------------------------------
// Kernel 1: per (b, group-pair, j) compute two 8x8 Gram matrices as one fp32
// 16x16 WMMA accumulation over L=190*190, then argmin / u_max / c_max.
// One block = 4 waves; each wave owns a slice of the 190 rows.
// Inner loop is branch-free (47 full K=4 chunks); the 2-column remainder is
// handled by one masked tail chunk per row.
// ---------------------------------------------------------------------------
__global__ __launch_bounds__(128) void motif_gram_kernel(
    const float* __restrict__ feature, int* __restrict__ umax,
    float* __restrict__ cmax) {
  __shared__ float sgram[4 * 256 + 256];
  __shared__ int sidx[16];

  const int blk = blockIdx.x;        // (b*8 + gp)*9 + j  -> 144 blocks
  const int j   = blk % 9;
  const int gp  = (blk / 9) % 8;     // group pair 0..7
  const int b   = blk / 72;
  const int ki  = j / 3, kj = j % 3;

  const int tid   = threadIdx.x;
  const int wave  = tid >> 5;
  const int lane  = tid & 31;
  const int r     = lane & 15;       // matrix row (= col, symmetric), 0..15
  const int khalf = lane >> 4;       // which K pair this lane feeds

  const int ch = gp * 16 + r;        // global channel for this matrix row
  const float* chbase = feature + (size_t)(b * Cn + ch) * Hn * Wn;

  v8f acc = {};

  const int ybeg = wave * 48;
  const int yend = (ybeg + 48 < Hm) ? (ybeg + 48) : Hm;

  for (int y = ybeg; y < yend; ++y) {
    const float* p = chbase + (size_t)(y + ki) * Wn + kj;

    // 47 full chunks: x = 0..187, xa+1 <= 187 < Wm -> no bounds checks.
    for (int x0 = 0; x0 < 188; x0 += 4) {
      const int xa = x0 + 2 * khalf;
      v2f a;
      a.x = p[xa];
      a.y = p[xa + 1];
      // Symmetric Gram: per-lane A and B register contents are identical.
      acc = __builtin_amdgcn_wmma_f32_16x16x4_f32(
          /*neg_a=*/false, a, /*neg_b=*/false, a,
          /*c_mod=*/(short)0, acc, /*reuse_a=*/false, /*reuse_b=*/false);
    }

    // Tail chunk: x = 188,189 live in K=0,1; K=2,3 lanes are zero padding.
    {
      v2f a;
      if (khalf == 0) {
        a.x = p[188];
        a.y = p[189];
      } else {
        a.x = 0.0f;
        a.y = 0.0f;
      }
      acc = __builtin_amdgcn_wmma_f32_16x16x4_f32(
          false, a, false, a, (short)0, acc, false, false);
    }
  }

  // Spill this wave's 16x16 accumulator: VGPR v holds D[M = v + 8*khalf][N=r].
#pragma unroll
  for (int v = 0; v < 8; ++v) {
    const int M = v + 8 * khalf;
    sgram[wave * 256 + M * 16 + r] = acc[v];
  }
  __syncthreads();

  // Reduce the 4 partial Grams -> sgram[1024..1279]
  for (int i = tid; i < 256; i += 128) {
    sgram[1024 + i] =
        sgram[i] + sgram[256 + i] + sgram[512 + i] + sgram[768 + i];
  }
  __syncthreads();

  const float* gm = &sgram[1024];
  if (tid < 16) {
    const int c    = tid;
    const int base = (c >> 3) * 8;           // this row's group block
    const float gcc = gm[c * 16 + c];
    float best = __builtin_inff();
    int bestd = 0;
    for (int d = 0; d < 8; ++d) {            // ascending + strict '<' == first-min (argmin)
      const int dd = base + d;
      if (dd == c) continue;                 // diagonal = inf
      const float v = gcc + gm[dd * 16 + dd] - 2.0f * gm[c * 16 + dd];
      if (v < best) { best = v; bestd = d; }
    }
    sidx[c] = bestd;                         // index local to the group (0..7)
  }
  __syncthreads();

  if (tid < 2) {                             // one thread per group in the pair
    int um = 0;
    for (int c = 0; c < 8; ++c) { int v = sidx[tid * 8 + c]; um = v > um ? v : um; }
    int cnt = 0;
    for (int c = 0; c < 8; ++c) cnt += (sidx[tid * 8 + c] == um);
    const int g = gp * 2 + tid;
    const int o = (b * Gn + g) * 9 + j;
    umax[o] = um;
    cmax[o] = (float)cnt;
  }
}

// ---------------------------------------------------------------------------
// Kernel 2: fused out_ser + Fold (gather form). For each output pixel, sum
// the <=9 fold taps; each tap re-derives its motif values straight from
// feature (L2-resident) using the scrambled (c',k') -> (j,c) remap:
//   flat = c'*9 + k'  ==  j*8 + c
// ---------------------------------------------------------------------------
__global__ __launch_bounds__(256) void motif_fold_kernel(
    const float* __restrict__ feature, const int* __restrict__ umax,
    const float* __restrict__ cmax, float* __restrict__ out) {
  const int idx = blockIdx.x * 256 + threadIdx.x;   // exact grid, no tail
  const int x  = idx % Wn;
  int t        = idx / Wn;
  const int y  = t % Hn;
  t /= Hn;
  const int ch = t % Cn;
  const int b  = t / Cn;

  const int g  = ch >> 3;
  const int cp = ch & 7;                            // c'
  const float bscale = (b != 0) ? 1.0f : 0.0f;      // l==0 branch drops motif

  const float* gbase = feature + (size_t)(b * Cn + g * CG) * Hn * Wn;
  const int*   up = umax + (b * Gn + g) * 9;
  const float* cm = cmax + (b * Gn + g) * 9;

  float sum = 0.0f;
#pragma unroll
  for (int kp = 0; kp < 9; ++kp) {
    const int kip = kp / 3, kjp = kp % 3;
    const int ym = y - kip, xm = x - kjp;
    if (ym < 0 || ym >= Hm || xm < 0 || xm >= Wm) continue;
    const int flat = cp * 9 + kp;
    const int jj = flat >> 3;                       // source (j, c) of this tap
    const int c  = flat & 7;
    const int yy = ym + jj / 3;
    const int xx = xm + jj % 3;
    const float a  = gbase[((size_t)c * Hn + yy) * Wn + xx];
    const float nn = gbase[((size_t)up[jj] * Hn + yy) * Wn + xx];
    sum += a * bscale + floorf(a * nn * cm[jj] * 0.125f);  // /Cg == *0.125 exact
  }
  out[idx] = sum;
}

// ---------------------------------------------------------------------------
extern "C" void kernel_launch(void* const* d_in, const int* in_sizes, int n_in,
                              void* d_out, int out_size, void* d_ws,
                              size_t ws_size, hipStream_t stream) {
  const float* feature = (const float*)d_in[0];
  float* out = (float*)d_out;

  int*   umax = (int*)d_ws;                                    // 288 ints
  float* cmax = (float*)((char*)d_ws + 288 * sizeof(int));     // 288 floats

  motif_gram_kernel<<<144, 128, 0, stream>>>(feature, umax, cmax);
  motif_fold_kernel<<<36864, 256, 0, stream>>>(feature, umax, cmax, out);
}